// ProbabilisticSurfaceDistanceLoss_7576322310561
// MI455X (gfx1250) — compile-verified
//
#include <hip/hip_runtime.h>
#include <hip/hip_bf16.h>

typedef float v2f __attribute__((ext_vector_type(2)));
typedef float v8f __attribute__((ext_vector_type(8)));

#define PENALTY_COEF 1e-4f
#define REVERSE_SCALE 0.1f
#define SENTINEL 1.0e30f
#define CHUNK_TILES 64                      // B tiles per wave (16 cols each)
#define COLS_PER_CHUNK (CHUNK_TILES * 16)   // 1024 database entries per chunk
#define OVERRUN_ROWS 64                     // 4 extra sentinel tiles for load-ahead

// Order-preserving float<->uint map: f1 < f2  <=>  fenc(f1) < fenc(f2) (uints).
// Makes atomicMin on floats deterministic (order-independent) for any sign.
__device__ __forceinline__ unsigned fenc(float f) {
    unsigned b = __float_as_uint(f);
    return (b & 0x80000000u) ? ~b : (b | 0x80000000u);
}
__device__ __forceinline__ float fdec(unsigned e) {
    unsigned b = (e & 0x80000000u) ? (e & 0x7FFFFFFFu) : ~e;
    return __uint_as_float(b);
}

// ---------------------------------------------------------------------------
// Prep kernels: build WMMA-ready operand arrays.
//   A rows:  [-2x, -2y, -2z, 1]   (queries), plus a2[row] = x^2+y^2+z^2
//   B rows:  [ bx,  by,  bz, b2]  (database, consumed as WMMA B columns)
// One V_WMMA_F32_16X16X4_F32 yields  b2 - 2 a.b  per (row,col) of a 16x16
// tile; min-dist(row) = a2[row] + min over cols. Database pad rows carry
// b2 = 1e30 so they never win the min.
// ---------------------------------------------------------------------------

__global__ void k_init(unsigned* __restrict__ enc_f, int nf,
                       unsigned* __restrict__ enc_r, int nr,
                       unsigned* __restrict__ max_bits) {
    int i = blockIdx.x * blockDim.x + threadIdx.x;
    if (i < nf) enc_f[i] = 0xFFFFFFFFu;   // +inf under fenc ordering
    if (i < nr) enc_r[i] = 0xFFFFFFFFu;
    if (i == 0) max_bits[0] = 0u;
}

__global__ void k_orig_bary_B(const float* __restrict__ ov, const int* __restrict__ of,
                              float* __restrict__ B4, int Fo, int FoAlloc) {
    int i = blockIdx.x * blockDim.x + threadIdx.x;
    if (i >= FoAlloc) return;
    if (i < Fo) {
        int i0 = of[3 * i + 0], i1 = of[3 * i + 1], i2 = of[3 * i + 2];
        float x = (ov[3 * i0 + 0] + ov[3 * i1 + 0] + ov[3 * i2 + 0]) * (1.0f / 3.0f);
        float y = (ov[3 * i0 + 1] + ov[3 * i1 + 1] + ov[3 * i2 + 1]) * (1.0f / 3.0f);
        float z = (ov[3 * i0 + 2] + ov[3 * i1 + 2] + ov[3 * i2 + 2]) * (1.0f / 3.0f);
        B4[4 * i + 0] = x; B4[4 * i + 1] = y; B4[4 * i + 2] = z;
        B4[4 * i + 3] = x * x + y * y + z * z;
    } else {
        B4[4 * i + 0] = 0.f; B4[4 * i + 1] = 0.f; B4[4 * i + 2] = 0.f;
        B4[4 * i + 3] = SENTINEL;
    }
}

__global__ void k_simp_bary_A(const float* __restrict__ sv, const int* __restrict__ sf,
                              float* __restrict__ A4, float* __restrict__ a2,
                              int Fs, int FsP) {
    int i = blockIdx.x * blockDim.x + threadIdx.x;
    if (i >= FsP) return;
    if (i < Fs) {
        int i0 = sf[3 * i + 0], i1 = sf[3 * i + 1], i2 = sf[3 * i + 2];
        float x = (sv[3 * i0 + 0] + sv[3 * i1 + 0] + sv[3 * i2 + 0]) * (1.0f / 3.0f);
        float y = (sv[3 * i0 + 1] + sv[3 * i1 + 1] + sv[3 * i2 + 1]) * (1.0f / 3.0f);
        float z = (sv[3 * i0 + 2] + sv[3 * i1 + 2] + sv[3 * i2 + 2]) * (1.0f / 3.0f);
        A4[4 * i + 0] = -2.f * x; A4[4 * i + 1] = -2.f * y;
        A4[4 * i + 2] = -2.f * z; A4[4 * i + 3] = 1.f;
        a2[i] = x * x + y * y + z * z;
    } else {
        A4[4 * i + 0] = 0.f; A4[4 * i + 1] = 0.f; A4[4 * i + 2] = 0.f; A4[4 * i + 3] = 0.f;
        a2[i] = 0.f;
    }
}

__global__ void k_samples_A(const float* __restrict__ sv, const int* __restrict__ sf,
                            const float* __restrict__ u1, const float* __restrict__ u2,
                            float* __restrict__ A4, float* __restrict__ p2,
                            int S, int Ns, int NsP) {
    int i = blockIdx.x * blockDim.x + threadIdx.x;
    if (i >= NsP) return;
    if (i < Ns) {
        int f = i / S;
        int i0 = sf[3 * f + 0], i1 = sf[3 * f + 1], i2 = sf[3 * f + 2];
        float r1 = sqrtf(u1[i]);
        float r2 = u2[i];
        float wa = 1.0f - r1;
        float wb = r1 * (1.0f - r2);
        float wc = r1 * r2;
        float x = wa * sv[3 * i0 + 0] + wb * sv[3 * i1 + 0] + wc * sv[3 * i2 + 0];
        float y = wa * sv[3 * i0 + 1] + wb * sv[3 * i1 + 1] + wc * sv[3 * i2 + 1];
        float z = wa * sv[3 * i0 + 2] + wb * sv[3 * i1 + 2] + wc * sv[3 * i2 + 2];
        A4[4 * i + 0] = -2.f * x; A4[4 * i + 1] = -2.f * y;
        A4[4 * i + 2] = -2.f * z; A4[4 * i + 3] = 1.f;
        p2[i] = x * x + y * y + z * z;
    } else {
        A4[4 * i + 0] = 0.f; A4[4 * i + 1] = 0.f; A4[4 * i + 2] = 0.f; A4[4 * i + 3] = 0.f;
        p2[i] = 0.f;
    }
}

__global__ void k_vert_B(const float* __restrict__ ov, float* __restrict__ B4,
                         int Vo, int VoAlloc) {
    int i = blockIdx.x * blockDim.x + threadIdx.x;
    if (i >= VoAlloc) return;
    if (i < Vo) {
        float x = ov[3 * i + 0], y = ov[3 * i + 1], z = ov[3 * i + 2];
        B4[4 * i + 0] = x; B4[4 * i + 1] = y; B4[4 * i + 2] = z;
        B4[4 * i + 3] = x * x + y * y + z * z;
    } else {
        B4[4 * i + 0] = 0.f; B4[4 * i + 1] = 0.f; B4[4 * i + 2] = 0.f;
        B4[4 * i + 3] = SENTINEL;
    }
}

// ---------------------------------------------------------------------------
// WMMA kNN-min kernel. ONE WAVE PER WORKGROUP (blockDim = 32): rowTile/chunk
// derive from blockIdx.x, so all control flow is scalar and EXEC stays all-1s
// through every WMMA (ISA requirement). The database is padded so every chunk
// is a full CHUNK_TILES tiles plus 4 overrun tiles of sentinel, making the
// loop branchless with a COMPILE-TIME trip count and a 4-deep load pipeline:
// 4 global_load_b64 stay in flight while 4 WMMAs + min chains execute.
//
// V_WMMA_F32_16X16X4_F32 layouts (ISA 7.12.2):
//   A 16x4 : lane<16 -> M=lane, regs {K0,K1}; lane>=16 -> M=lane-16, {K2,K3}
//   B 4x16 : lane<16 -> N=lane, regs {K0,K1}; lane>=16 -> N=lane-16, {K2,K3}
//   D 16x16: reg v, lanes 0-15 -> (M=v, N=lane); lanes 16-31 -> (M=v+8, N=lane-16)
// ---------------------------------------------------------------------------
__global__ void __launch_bounds__(32)
k_knn_min_wmma(const float* __restrict__ A4,
               const float* __restrict__ B4,
               unsigned* __restrict__ minenc,
               int nChunks, int Mreal) {
    // Scalar (SGPR) work assignment: uniform by construction.
    const int rowTile = blockIdx.x / nChunks;
    const int chunk   = blockIdx.x - rowTile * nChunks;

    const int lane = threadIdx.x;     // blockDim == 32
    const int half = lane >> 4;       // 0: regs hold K0,K1 ; 1: K2,K3
    const int l16  = lane & 15;

    v2f a = *(const v2f*)(A4 + (size_t)(rowTile * 16 + l16) * 4 + half * 2);

    float m0 = 3.0e38f, m1 = 3.0e38f, m2 = 3.0e38f, m3 = 3.0e38f;
    float m4 = 3.0e38f, m5 = 3.0e38f, m6 = 3.0e38f, m7 = 3.0e38f;

    // Per-lane column base inside this chunk (64 floats per 16-col tile).
    const float* bp = B4 + (size_t)chunk * (CHUNK_TILES * 64)
                         + (size_t)l16 * 4 + half * 2;

    // 4-deep software pipeline (overrun tiles are valid sentinel memory).
    v2f b0 = *(const v2f*)(bp + 0 * 64);
    v2f b1 = *(const v2f*)(bp + 1 * 64);
    v2f b2 = *(const v2f*)(bp + 2 * 64);
    v2f b3 = *(const v2f*)(bp + 3 * 64);

#define TILE_MIN(B)                                                          \
    do {                                                                     \
        v8f c = {};                                                          \
        v8f d = __builtin_amdgcn_wmma_f32_16x16x4_f32(                       \
            false, a, false, (B), (short)0, c, false, false);                \
        m0 = fminf(m0, d[0]); m1 = fminf(m1, d[1]);                          \
        m2 = fminf(m2, d[2]); m3 = fminf(m3, d[3]);                          \
        m4 = fminf(m4, d[4]); m5 = fminf(m5, d[5]);                          \
        m6 = fminf(m6, d[6]); m7 = fminf(m7, d[7]);                          \
    } while (0)

    for (int t = 0; t < CHUNK_TILES; t += 4) {     // static trip count (16)
        v2f n0 = *(const v2f*)(bp + (size_t)(t + 4) * 64);
        v2f n1 = *(const v2f*)(bp + (size_t)(t + 5) * 64);
        v2f n2 = *(const v2f*)(bp + (size_t)(t + 6) * 64);
        v2f n3 = *(const v2f*)(bp + (size_t)(t + 7) * 64);
        __builtin_prefetch(bp + (size_t)(t + 8) * 64, 0, 0);  // global_prefetch_b8
        TILE_MIN(b0);
        TILE_MIN(b1);
        TILE_MIN(b2);
        TILE_MIN(b3);
        b0 = n0; b1 = n1; b2 = n2; b3 = n3;
    }
#undef TILE_MIN

    // Min over the 16-lane halves (xor masks 1..8 stay within a half).
    float m[8] = {m0, m1, m2, m3, m4, m5, m6, m7};
    #pragma unroll
    for (int off = 1; off < 16; off <<= 1) {
        #pragma unroll
        for (int v = 0; v < 8; ++v)
            m[v] = fminf(m[v], __shfl_xor(m[v], off, 32));
    }

    // lane 0 -> rows rowTile*16+0..7, lane 16 -> rows rowTile*16+8..15
    if (l16 == 0) {
        const int base = rowTile * 16 + half * 8;
        #pragma unroll
        for (int v = 0; v < 8; ++v) {
            int r = base + v;
            if (r < Mreal) atomicMin(&minenc[r], fenc(m[v]));
        }
    }
}

// Reverse epilogue: nn_d2[i] = p2[i] + min; global max via order-independent
// atomicMax on non-negative float bits (block pre-reduction first).
__global__ void k_nn_max(const float* __restrict__ p2, const unsigned* __restrict__ enc_r,
                         float* __restrict__ nn, unsigned* __restrict__ max_bits, int Ns) {
    __shared__ float red[256];
    int i = blockIdx.x * blockDim.x + threadIdx.x;
    float v = 0.f;
    if (i < Ns) {
        float d = p2[i] + fdec(enc_r[i]);
        nn[i] = d;
        v = fmaxf(d, 0.f);
    }
    red[threadIdx.x] = v;
    __syncthreads();
    for (int o = blockDim.x >> 1; o > 0; o >>= 1) {
        if ((int)threadIdx.x < o) red[threadIdx.x] = fmaxf(red[threadIdx.x], red[threadIdx.x + o]);
        __syncthreads();
    }
    if (threadIdx.x == 0) atomicMax(max_bits, __float_as_uint(red[0]));
}

// Final scalar (single block, deterministic summation order).
__global__ void k_finalize(const float* __restrict__ prob, const float* __restrict__ nn,
                           const float* __restrict__ a2f, const unsigned* __restrict__ enc_f,
                           const unsigned* __restrict__ max_bits,
                           float* __restrict__ out, int Fs, int S, int Ns) {
    __shared__ float red[256];
    const float maxv = __uint_as_float(*max_bits);
    const float inv  = (maxv > 0.f) ? (REVERSE_SCALE / maxv) : 0.f;

    float s = 0.f;
    for (int i = threadIdx.x; i < Ns; i += blockDim.x)
        s += prob[i / S] * nn[i] * inv;                       // reverse term
    for (int f = threadIdx.x; f < Fs; f += blockDim.x)
        s += prob[f] * (a2f[f] + fdec(enc_f[f]))              // forward term
           + PENALTY_COEF * (1.0f - prob[f]);                 // penalty

    red[threadIdx.x] = s;
    __syncthreads();
    for (int o = blockDim.x >> 1; o > 0; o >>= 1) {
        if ((int)threadIdx.x < o) red[threadIdx.x] += red[threadIdx.x + o];
        __syncthreads();
    }
    if (threadIdx.x == 0) out[0] = red[0];
}

// ---------------------------------------------------------------------------
extern "C" void kernel_launch(void* const* d_in, const int* in_sizes, int n_in,
                              void* d_out, int out_size, void* d_ws, size_t ws_size,
                              hipStream_t stream) {
    const float* ov   = (const float*)d_in[0];  // original_vertices  [Vo,3]
    const int*   of   = (const int*)  d_in[1];  // original_faces     [Fo,3]
    const float* sv   = (const float*)d_in[2];  // simplified_vertices[Vs,3]
    const int*   sf   = (const int*)  d_in[3];  // simplified_faces   [Fs,3]
    const float* prob = (const float*)d_in[4];  // face_probabilities [Fs]
    const float* u1   = (const float*)d_in[5];  // rand_u1            [Fs,S,1]
    const float* u2   = (const float*)d_in[6];  // rand_u2            [Fs,S,1]

    const int Vo = in_sizes[0] / 3;
    const int Fo = in_sizes[1] / 3;
    const int Fs = in_sizes[3] / 3;
    const int S  = in_sizes[5] / Fs;
    const int Ns = Fs * S;

    auto pad16 = [](int x) { return (x + 15) & ~15; };
    auto padDB = [](int x) {  // database: whole chunks
        return ((x + COLS_PER_CHUNK - 1) / COLS_PER_CHUNK) * COLS_PER_CHUNK;
    };
    const int FsP = pad16(Fs), NsP = pad16(Ns);
    const int FoDB = padDB(Fo), VoDB = padDB(Vo);
    const int FoAlloc = FoDB + OVERRUN_ROWS;   // + load-ahead sentinel tiles
    const int VoAlloc = VoDB + OVERRUN_ROWS;

    float* w = (float*)d_ws;
    float* Afwd = w; w += (size_t)FsP * 4;
    float* a2f  = w; w += FsP;
    float* Bfwd = w; w += (size_t)FoAlloc * 4;
    float* Asmp = w; w += (size_t)NsP * 4;
    float* p2s  = w; w += NsP;
    float* Brev = w; w += (size_t)VoAlloc * 4;
    float* nn   = w; w += NsP;
    unsigned* encF = (unsigned*)w; w += FsP;
    unsigned* encR = (unsigned*)w; w += NsP;
    unsigned* maxb = (unsigned*)w;

    const int B = 256;
    const int initN = (FsP > NsP ? FsP : NsP);
    k_init      <<<(initN + B - 1) / B, B, 0, stream>>>(encF, FsP, encR, NsP, maxb);
    k_orig_bary_B<<<(FoAlloc + B - 1) / B, B, 0, stream>>>(ov, of, Bfwd, Fo, FoAlloc);
    k_simp_bary_A<<<(FsP + B - 1) / B, B, 0, stream>>>(sv, sf, Afwd, a2f, Fs, FsP);
    k_samples_A <<<(NsP + B - 1) / B, B, 0, stream>>>(sv, sf, u1, u2, Asmp, p2s, S, Ns, NsP);
    k_vert_B    <<<(VoAlloc + B - 1) / B, B, 0, stream>>>(ov, Brev, Vo, VoAlloc);

    // Forward: Fs queries vs Fo barycenters. N split into full chunks.
    {
        const int rowTiles = FsP / 16;
        const int nChunks  = FoDB / COLS_PER_CHUNK;
        k_knn_min_wmma<<<rowTiles * nChunks, 32, 0, stream>>>(
            Afwd, Bfwd, encF, nChunks, Fs);
    }
    // Reverse: Ns sample points vs Vo vertices.
    {
        const int rowTiles = NsP / 16;
        const int nChunks  = VoDB / COLS_PER_CHUNK;
        k_knn_min_wmma<<<rowTiles * nChunks, 32, 0, stream>>>(
            Asmp, Brev, encR, nChunks, Ns);
    }

    k_nn_max  <<<(Ns + B - 1) / B, B, 0, stream>>>(p2s, encR, nn, maxb, Ns);
    k_finalize<<<1, 256, 0, stream>>>(prob, nn, a2f, encF, maxb,
                                      (float*)d_out, Fs, S, Ns);
}